// GNN_53532472378064
// MI455X (gfx1250) — compile-verified
//
#include <hip/hip_runtime.h>
#include <hip/hip_bf16.h>

typedef __attribute__((ext_vector_type(2))) float v2f;
typedef __attribute__((ext_vector_type(8))) float v8f;

#define DD 64
#define EPS 1e-5f

// ---------------- degree / norm prep ----------------
__global__ void k_deg_init(float* deg, int n) {
    int i = blockIdx.x * blockDim.x + threadIdx.x;
    if (i < n) deg[i] = 1.0f;   // self loop
}

__global__ void k_deg_accum(const int* __restrict__ dst, float* deg, int e) {
    int i = blockIdx.x * blockDim.x + threadIdx.x;
    if (i < e) unsafeAtomicAdd(&deg[dst[i]], 1.0f);
}

__global__ void k_deg_to_dis(float* deg, int n) {
    int i = blockIdx.x * blockDim.x + threadIdx.x;
    if (i < n) deg[i] = rsqrtf(deg[i]);   // dis = deg^{-1/2}
}

__global__ void k_edge_norm(const int* __restrict__ src, const int* __restrict__ dst,
                            const float* __restrict__ dis, float* __restrict__ enorm, int e) {
    int i = blockIdx.x * blockDim.x + threadIdx.x;
    if (i < e) enorm[i] = dis[src[i]] * dis[dst[i]];
}

// ---------------- fp32 WMMA GEMM: C[M x 64] = A[M x K] * W[K x 64] ----------------
// block = 128 threads (4 waves). Each wave computes a 16x64 slab: one A fragment
// per k-step feeds 4 independent V_WMMA_F32_16X16X4_F32 accumulators (4 col tiles).
// Fragment layouts per CDNA5 ISA 7.12.2.
__global__ void k_gemm_wmma(const float* __restrict__ A, const float* __restrict__ W,
                            float* __restrict__ C, int M, int K) {
    const int wave = threadIdx.x >> 5;              // 0..3 -> row group
    const int lane = threadIdx.x & 31;
    const int half = lane >> 4;                     // 0/1
    const int ln   = lane & 15;                     // 0..15
    const int r0   = blockIdx.x * 64 + wave * 16;
    if (r0 >= M) return;                            // wave-uniform exit, EXEC all-ones

    v8f acc0 = {}, acc1 = {}, acc2 = {}, acc3 = {};
    // A 16x4 f32 fragment: v0 holds K=2*half, v1 holds K=2*half+1 (row = r0+ln)
    const float* __restrict__ arow  = A + (size_t)(r0 + ln) * K + 2 * half;
    // B 4x16 f32 fragment: rows K=2*half / K=2*half+1, columns striped on lanes
    const float* __restrict__ wbase = W + (size_t)(2 * half) * DD + ln;

    for (int k0 = 0; k0 < K; k0 += 4) {
        v2f a, b0, b1, b2, b3;
        a.x = arow[k0];
        a.y = arow[k0 + 1];
        const float* __restrict__ wr = wbase + (size_t)k0 * DD;
        b0.x = wr[0];       b0.y = wr[0 + DD];
        b1.x = wr[16];      b1.y = wr[16 + DD];
        b2.x = wr[32];      b2.y = wr[32 + DD];
        b3.x = wr[48];      b3.y = wr[48 + DD];
        acc0 = __builtin_amdgcn_wmma_f32_16x16x4_f32(false, a, false, b0, (short)0, acc0, false, false);
        acc1 = __builtin_amdgcn_wmma_f32_16x16x4_f32(false, a, false, b1, (short)0, acc1, false, false);
        acc2 = __builtin_amdgcn_wmma_f32_16x16x4_f32(false, a, false, b2, (short)0, acc2, false, false);
        acc3 = __builtin_amdgcn_wmma_f32_16x16x4_f32(false, a, false, b3, (short)0, acc3, false, false);
    }
    // C 16x16 f32: VGPR r -> row r (lanes 0-15) / row r+8 (lanes 16-31), col = ln
    float* crow = C + (size_t)(r0 + 8 * half) * DD + ln;
#pragma unroll
    for (int r = 0; r < 8; ++r) {
        crow[(size_t)r * DD + 0]  = acc0[r];
        crow[(size_t)r * DD + 16] = acc1[r];
        crow[(size_t)r * DD + 32] = acc2[r];
        crow[(size_t)r * DD + 48] = acc3[r];
    }
}

// ---------------- agg = hw * self_norm + bias (float4, b128 path) ----------------
__global__ void k_agg_init(const float4* __restrict__ hw, const float* __restrict__ dis,
                           const float4* __restrict__ b, float4* __restrict__ agg, int n) {
    int t = blockIdx.x * blockDim.x + threadIdx.x;      // over n*16 float4s
    if (t < n * (DD / 4)) {
        int i = t >> 4, d4 = t & 15;
        float s = dis[i]; s *= s;
        float4 h = hw[t], bb = b[d4];
        float4 o;
        o.x = h.x * s + bb.x; o.y = h.y * s + bb.y;
        o.z = h.z * s + bb.z; o.w = h.w * s + bb.w;
        agg[t] = o;
    }
}

// ---------------- edge scatter: agg[dst] += hw[src] * enorm ----------------
__global__ void k_edge_scatter(const int* __restrict__ src, const int* __restrict__ dst,
                               const float* __restrict__ enorm, const float* __restrict__ hw,
                               float* __restrict__ agg, int e) {
    long long t = (long long)blockIdx.x * blockDim.x + threadIdx.x;
    if (t < (long long)e * DD) {
        int ei = (int)(t >> 6), d = (int)(t & 63);
        unsafeAtomicAdd(&agg[(size_t)dst[ei] * DD + d], hw[(size_t)src[ei] * DD + d] * enorm[ei]);
    }
}

// ---------------- BN stats: per-column sum & sumsq ----------------
__global__ void k_bn_stats(const float* __restrict__ agg, float* __restrict__ stats, int n) {
    __shared__ float s1[256], s2[256];
    int d = threadIdx.x & 63;
    int rg = threadIdx.x >> 6;                  // 0..3
    float a1 = 0.0f, a2 = 0.0f;
    for (int r = blockIdx.x * 4 + rg; r < n; r += gridDim.x * 4) {
        float v = agg[(size_t)r * DD + d];
        a1 += v; a2 += v * v;
    }
    s1[threadIdx.x] = a1; s2[threadIdx.x] = a2;
    __syncthreads();
    if (threadIdx.x < 64) {
        float t1 = s1[d] + s1[d + 64] + s1[d + 128] + s1[d + 192];
        float t2 = s2[d] + s2[d + 64] + s2[d + 128] + s2[d + 192];
        unsafeAtomicAdd(&stats[d], t1);
        unsafeAtomicAdd(&stats[DD + d], t2);
    }
}

// ---------------- BN apply (+ optional ReLU), float4 ----------------
__global__ void k_bn_apply(const float4* __restrict__ agg, const float4* __restrict__ stats,
                           const float4* __restrict__ g, const float4* __restrict__ bt,
                           float4* __restrict__ out, int n, int relu) {
    int t = blockIdx.x * blockDim.x + threadIdx.x;      // over n*16 float4s
    if (t < n * (DD / 4)) {
        int d4 = t & 15;
        float inv_n = 1.0f / (float)n;
        float4 s1 = stats[d4];                 // column sums
        float4 s2 = stats[16 + d4];            // column sumsq
        float4 gg = g[d4], bb = bt[d4], v = agg[t];
        float4 o;
        {
            float mu = s1.x * inv_n, var = s2.x * inv_n - mu * mu;
            o.x = (v.x - mu) * rsqrtf(var + EPS) * gg.x + bb.x;
        }
        {
            float mu = s1.y * inv_n, var = s2.y * inv_n - mu * mu;
            o.y = (v.y - mu) * rsqrtf(var + EPS) * gg.y + bb.y;
        }
        {
            float mu = s1.z * inv_n, var = s2.z * inv_n - mu * mu;
            o.z = (v.z - mu) * rsqrtf(var + EPS) * gg.z + bb.z;
        }
        {
            float mu = s1.w * inv_n, var = s2.w * inv_n - mu * mu;
            o.w = (v.w - mu) * rsqrtf(var + EPS) * gg.w + bb.w;
        }
        if (relu) {
            o.x = fmaxf(o.x, 0.0f); o.y = fmaxf(o.y, 0.0f);
            o.z = fmaxf(o.z, 0.0f); o.w = fmaxf(o.w, 0.0f);
        }
        out[t] = o;
    }
}

// ---------------- global add pool ----------------
__global__ void k_pool(const float* __restrict__ h, const int* __restrict__ batch,
                       float* __restrict__ pool, int n) {
    int t = blockIdx.x * blockDim.x + threadIdx.x;
    if (t < n * DD) {
        int i = t >> 6, d = t & 63;
        unsafeAtomicAdd(&pool[(size_t)batch[i] * DD + d], h[t]);
    }
}

extern "C" void kernel_launch(void* const* d_in, const int* in_sizes, int n_in,
                              void* d_out, int out_size, void* d_ws, size_t ws_size,
                              hipStream_t stream) {
    const float* x       = (const float*)d_in[0];
    const int*   eidx    = (const int*)  d_in[1];
    const int*   batch   = (const int*)  d_in[2];
    const float* W_embed = (const float*)d_in[3];
    const float* W1      = (const float*)d_in[4];
    const float* b1      = (const float*)d_in[5];
    const float* g1      = (const float*)d_in[6];
    const float* bt1     = (const float*)d_in[7];
    const float* W2      = (const float*)d_in[8];
    const float* b2      = (const float*)d_in[9];
    const float* g2      = (const float*)d_in[10];
    const float* bt2     = (const float*)d_in[11];

    const int N  = in_sizes[2];            // batch vector is length N
    const int IN = in_sizes[0] / N;        // 32
    const int E  = in_sizes[1] / 2;
    const int G  = out_size / DD - N;      // pool rows

    const int* src = eidx;                 // edge_index[0]
    const int* dst = eidx + E;             // edge_index[1]

    // workspace carve-up (floats)
    float* ws    = (float*)d_ws;
    float* dis   = ws;                     // [N]
    float* enorm = dis + N;                // [E]
    float* stats = enorm + E;              // [2*DD] (16B aligned: N,E sizes are multiples of 4)
    float* hcur  = stats + 2 * DD;         // [N*DD]
    float* hw    = hcur + (size_t)N * DD;  // [N*DD]
    float* agg   = hw   + (size_t)N * DD;  // [N*DD]

    float* h_out    = (float*)d_out;             // [N*DD]
    float* pool_out = h_out + (size_t)N * DD;    // [G*DD]

    const int NT = 256;
    const int gN   = (N + NT - 1) / NT;
    const int gE   = (E + NT - 1) / NT;
    const int gND  = (int)(((long long)N * DD + NT - 1) / NT);       // scalar n*64
    const int gND4 = (int)(((long long)N * (DD / 4) + NT - 1) / NT); // float4 n*16
    const int gED  = (int)(((long long)E * DD + NT - 1) / NT);
    const int gGem = (N + 63) / 64;        // 64 rows per block (4 waves x 16)

    // degrees -> dis -> per-edge norms
    k_deg_init   <<<gN, NT, 0, stream>>>(dis, N);
    k_deg_accum  <<<gE, NT, 0, stream>>>(dst, dis, E);
    k_deg_to_dis <<<gN, NT, 0, stream>>>(dis, N);
    k_edge_norm  <<<gE, NT, 0, stream>>>(src, dst, dis, enorm, E);

    // embed: hcur = x @ W_embed   (WMMA fp32)
    k_gemm_wmma<<<gGem, 128, 0, stream>>>(x, W_embed, hcur, N, IN);

    // ---- layer 1 ----
    k_gemm_wmma<<<gGem, 128, 0, stream>>>(hcur, W1, hw, N, DD);
    k_agg_init<<<gND4, NT, 0, stream>>>((const float4*)hw, dis, (const float4*)b1,
                                        (float4*)agg, N);
    k_edge_scatter<<<gED, NT, 0, stream>>>(src, dst, enorm, hw, agg, E);
    hipMemsetAsync(stats, 0, 2 * DD * sizeof(float), stream);
    k_bn_stats<<<1024, NT, 0, stream>>>(agg, stats, N);
    k_bn_apply<<<gND4, NT, 0, stream>>>((const float4*)agg, (const float4*)stats,
                                        (const float4*)g1, (const float4*)bt1,
                                        (float4*)hcur, N, /*relu=*/1);

    // ---- layer 2 ----
    k_gemm_wmma<<<gGem, 128, 0, stream>>>(hcur, W2, hw, N, DD);
    k_agg_init<<<gND4, NT, 0, stream>>>((const float4*)hw, dis, (const float4*)b2,
                                        (float4*)agg, N);
    k_edge_scatter<<<gED, NT, 0, stream>>>(src, dst, enorm, hw, agg, E);
    hipMemsetAsync(stats, 0, 2 * DD * sizeof(float), stream);
    k_bn_stats<<<1024, NT, 0, stream>>>(agg, stats, N);
    k_bn_apply<<<gND4, NT, 0, stream>>>((const float4*)agg, (const float4*)stats,
                                        (const float4*)g2, (const float4*)bt2,
                                        (float4*)h_out, N, /*relu=*/0);

    // ---- pool ----
    hipMemsetAsync(pool_out, 0, (size_t)G * DD * sizeof(float), stream);
    k_pool<<<gND, NT, 0, stream>>>(h_out, batch, pool_out, N);
}